// DotProductSelfAttention_15410342658404
// MI455X (gfx1250) — compile-verified
//
#include <hip/hip_runtime.h>
#include <hip/hip_bf16.h>
#include <math.h>

#define B_SZ 32
#define T_SZ 4096
#define D_SZ 1024
#define K2D  2048   // 2*D
#define NCH  8      // t-chunks for deterministic context partials

typedef __attribute__((ext_vector_type(2))) float v2f;
typedef __attribute__((ext_vector_type(4))) float v4f;
typedef __attribute__((ext_vector_type(8))) float v8f;

// ---------------------------------------------------------------------------
// K1: scores[b,t] = dot(prev_refined[b,t,:], state_t[b,:])
// One wave32 per (b,t) row; state_t[b] staged in LDS; float4 streaming loads.
// prev_refined is a 512MB single-use stream (> 192MB L2) -> nontemporal loads.
// ---------------------------------------------------------------------------
__global__ void scores_kernel(const float* __restrict__ state,
                              const float* __restrict__ prev_ref,
                              float* __restrict__ scores) {
  __shared__ float s_state[D_SZ];
  const int b   = blockIdx.y;
  const int tid = threadIdx.x;
  // stage state_t[b] (1024 floats) into LDS: 256 threads x 1 float4
  ((v4f*)s_state)[tid] = ((const v4f*)(state + (size_t)b * D_SZ))[tid];
  __syncthreads();

  const int wave = tid >> 5;
  const int lane = tid & 31;
  const int t = blockIdx.x * 8 + wave;

  const v4f* row = (const v4f*)(prev_ref + ((size_t)b * T_SZ + t) * D_SZ);
  const v4f* st4 = (const v4f*)s_state;

  float acc = 0.f;
#pragma unroll
  for (int i = 0; i < 8; ++i) {
    const int idx = lane + 32 * i;                 // coalesced across the wave
    v4f p = __builtin_nontemporal_load(row + idx); // single-use stream: NT hint
    v4f s = st4[idx];
    acc += p.x * s.x + p.y * s.y + p.z * s.z + p.w * s.w;
  }
#pragma unroll
  for (int off = 16; off > 0; off >>= 1)
    acc += __shfl_xor(acc, off, 32);
  if (lane == 0) scores[(size_t)b * T_SZ + t] = acc;
}

// ---------------------------------------------------------------------------
// K2: softmax over T per batch row; writes attn_weights into d_out slice.
// ---------------------------------------------------------------------------
__global__ void softmax_kernel(const float* __restrict__ scores,
                               float* __restrict__ attn) {
  __shared__ float red[256];
  const int b   = blockIdx.x;
  const int tid = threadIdx.x;
  const float* srow = scores + (size_t)b * T_SZ;

  float vals[16];
  float mx = -INFINITY;
#pragma unroll
  for (int i = 0; i < 16; ++i) {
    vals[i] = srow[tid + 256 * i];
    mx = fmaxf(mx, vals[i]);
  }
  red[tid] = mx;
  __syncthreads();
  for (int s = 128; s > 0; s >>= 1) {
    if (tid < s) red[tid] = fmaxf(red[tid], red[tid + s]);
    __syncthreads();
  }
  mx = red[0];
  __syncthreads();

  float sum = 0.f;
#pragma unroll
  for (int i = 0; i < 16; ++i) {
    vals[i] = expf(vals[i] - mx);
    sum += vals[i];
  }
  red[tid] = sum;
  __syncthreads();
  for (int s = 128; s > 0; s >>= 1) {
    if (tid < s) red[tid] += red[tid + s];
    __syncthreads();
  }
  const float inv = 1.f / red[0];
#pragma unroll
  for (int i = 0; i < 16; ++i)
    attn[(size_t)b * T_SZ + tid + 256 * i] = vals[i] * inv;
}

// ---------------------------------------------------------------------------
// K3: context partials. Grid (D/256, NCH, B). Each block streams 512 rows of
// prev_states[b, t, d-slice] with attn chunk staged in LDS. Fixed-order sums
// per chunk keep the result deterministic (no float atomics).
// prev_states is the second 512MB single-use stream -> nontemporal loads.
// ---------------------------------------------------------------------------
__global__ void context_partial_kernel(const float* __restrict__ prev_states,
                                       const float* __restrict__ attn,
                                       float* __restrict__ partial) {
  __shared__ float s_attn[T_SZ / NCH];   // 512 floats
  const int b     = blockIdx.z;
  const int chunk = blockIdx.y;
  const int tid   = threadIdx.x;
  const int d     = blockIdx.x * 256 + tid;
  const int t0    = chunk * (T_SZ / NCH);

#pragma unroll
  for (int i = 0; i < (T_SZ / NCH) / 256; ++i)
    s_attn[tid + 256 * i] = attn[(size_t)b * T_SZ + t0 + tid + 256 * i];
  __syncthreads();

  float acc = 0.f;
  const float* base = prev_states + ((size_t)b * T_SZ + t0) * D_SZ + d;
  for (int tt = 0; tt < T_SZ / NCH; ++tt)
    acc += s_attn[tt] * __builtin_nontemporal_load(base + (size_t)tt * D_SZ);
  partial[((size_t)chunk * B_SZ + b) * D_SZ + d] = acc;
}

// ---------------------------------------------------------------------------
// K4: reduce partials and build fused A matrix [B, 2D] = [state | context]
// ---------------------------------------------------------------------------
__global__ void fuse_prep_kernel(const float* __restrict__ state,
                                 const float* __restrict__ partial,
                                 float* __restrict__ fusedA) {
  const int n = blockIdx.x * 256 + threadIdx.x;   // over B*2048
  const int b = n >> 11;
  const int k = n & (K2D - 1);
  float v;
  if (k < D_SZ) {
    v = state[b * D_SZ + k];
  } else {
    const int d = k - D_SZ;
    float acc = 0.f;
#pragma unroll
    for (int c = 0; c < NCH; ++c)
      acc += partial[((size_t)c * B_SZ + b) * D_SZ + d];
    v = acc;
  }
  fusedA[n] = v;
}

// ---------------------------------------------------------------------------
// K5: refined = tanh(fusedA @ W^T + bias) via V_WMMA_F32_16X16X4_F32.
// One wave per 16x16 output tile; 128 tiles total (2 M-tiles x 64 N-tiles).
// 32-bit A 16x4 layout: lanes 0-15 hold K=kk+{0,1}, lanes 16-31 K=kk+{2,3},
// M = lane&15 -> a contiguous float2 per lane. B[k][n] = W[n][k] gives the
// mirrored contiguous float2 from W's row n0+(lane&15).
// C/D layout: VGPR j -> row m0 + j + 8*(lane>=16), col n0 + (lane&15).
// ---------------------------------------------------------------------------
__global__ void fuse_wmma_kernel(const float* __restrict__ fusedA,
                                 const float* __restrict__ W,
                                 const float* __restrict__ bias,
                                 float* __restrict__ out) {
  const int lane  = threadIdx.x & 31;
  const int wib   = threadIdx.x >> 5;
  const int tile  = blockIdx.x * 4 + wib;       // 0..127
  const int tileN = tile & 63;
  const int tileM = tile >> 6;
  const int m0 = tileM * 16;
  const int n0 = tileN * 16;
  const int l15 = lane & 15;
  const int lhi = lane >> 4;

  const float* aptr = fusedA + (size_t)(m0 + l15) * K2D + 2 * lhi;
  const float* bptr = W      + (size_t)(n0 + l15) * K2D + 2 * lhi;

  v8f c = {0.f, 0.f, 0.f, 0.f, 0.f, 0.f, 0.f, 0.f};
  for (int kk = 0; kk < K2D; kk += 4) {
    v2f a  = *(const v2f*)(aptr + kk);
    v2f bb = *(const v2f*)(bptr + kk);
    c = __builtin_amdgcn_wmma_f32_16x16x4_f32(
        /*neg_a=*/false, a, /*neg_b=*/false, bb,
        /*c_mod=*/(short)0, c, /*reuse_a=*/false, /*reuse_b=*/false);
  }

  const float bv  = bias[n0 + l15];
  const int col   = n0 + l15;
  const int rbase = m0 + 8 * lhi;
#pragma unroll
  for (int j = 0; j < 8; ++j)
    out[(size_t)(rbase + j) * D_SZ + col] = tanhf(c[j] + bv);
}

// ---------------------------------------------------------------------------
extern "C" void kernel_launch(void* const* d_in, const int* in_sizes, int n_in,
                              void* d_out, int out_size, void* d_ws, size_t ws_size,
                              hipStream_t stream) {
  const float* state = (const float*)d_in[0];   // [B, D]
  const float* prev  = (const float*)d_in[1];   // [B, T, D]
  const float* prevr = (const float*)d_in[2];   // [B, T, D]
  const float* Wf    = (const float*)d_in[3];   // [D, 2D]
  const float* bf    = (const float*)d_in[4];   // [D]

  float* out     = (float*)d_out;
  float* refined = out;                         // [B, D]
  float* attn    = out + (size_t)B_SZ * D_SZ;   // [B, T]

  float* scores  = (float*)d_ws;                            // B*T
  float* partial = scores  + (size_t)B_SZ * T_SZ;           // NCH*B*D
  float* fusedA  = partial + (size_t)NCH * B_SZ * D_SZ;     // B*2D

  scores_kernel<<<dim3(T_SZ / 8, B_SZ), 256, 0, stream>>>(state, prevr, scores);
  softmax_kernel<<<B_SZ, 256, 0, stream>>>(scores, attn);
  context_partial_kernel<<<dim3(D_SZ / 256, NCH, B_SZ), 256, 0, stream>>>(prev, attn, partial);
  fuse_prep_kernel<<<(B_SZ * K2D) / 256, 256, 0, stream>>>(state, partial, fusedA);
  fuse_wmma_kernel<<<32, 128, 0, stream>>>(fusedA, Wf, bf, refined);
}